// DANet_88210038326168
// MI455X (gfx1250) — compile-verified
//
#include <hip/hip_runtime.h>
#include <math.h>

typedef __attribute__((ext_vector_type(2))) float v2f;
typedef __attribute__((ext_vector_type(8))) float v8f;

#define BN_EPS 1e-5f

// D = A(16x4, f32) * B(4x16, f32) + C(16x16, f32)
static __device__ __forceinline__ v8f wmma4(v2f a, v2f b, v8f c) {
  return __builtin_amdgcn_wmma_f32_16x16x4_f32(false, a, false, b, (short)0, c,
                                               false, false);
}

// ---------------------------------------------------------------------------
// Kernel 1: psi = relu(BN(Wg*g) + BN(Wx*x))   [B,64,4096]
// one wave = one 16x16 tile, K=256. tiles = 8*256*4 = 8192 -> 1024 blocks x 8 waves
// ---------------------------------------------------------------------------
__global__ void k_psi(const float* __restrict__ g, const float* __restrict__ x,
                      const float* __restrict__ Wg, const float* __restrict__ Wgb,
                      const float* __restrict__ gg, const float* __restrict__ gbt,
                      const float* __restrict__ gmu, const float* __restrict__ gvar,
                      const float* __restrict__ Wx, const float* __restrict__ Wxb,
                      const float* __restrict__ xg, const float* __restrict__ xbt,
                      const float* __restrict__ xmu, const float* __restrict__ xvar,
                      float* __restrict__ psi) {
  const int lane = threadIdx.x & 31, half = lane >> 4, lr = lane & 15, bk = half * 2;
  const int wid = blockIdx.x * 8 + (threadIdx.x >> 5);
  const int b = wid >> 10;
  const int rem = wid & 1023;
  const int nt = rem >> 2, mt = rem & 3;
  const int m0 = mt * 16, n0 = nt * 16;
  const float* gB = g + (size_t)b * 256 * 4096;
  const float* xB = x + (size_t)b * 256 * 4096;
  v8f ag = {}, ax = {};
  for (int k = 0; k < 256; k += 4) {
    v2f aAg = *(const v2f*)(Wg + (m0 + lr) * 256 + k + bk);
    v2f aAx = *(const v2f*)(Wx + (m0 + lr) * 256 + k + bk);
    v2f bBg, bBx;
    bBg.x = gB[(size_t)(k + bk) * 4096 + n0 + lr];
    bBg.y = gB[(size_t)(k + bk + 1) * 4096 + n0 + lr];
    bBx.x = xB[(size_t)(k + bk) * 4096 + n0 + lr];
    bBx.y = xB[(size_t)(k + bk + 1) * 4096 + n0 + lr];
    ag = wmma4(aAg, bBg, ag);
    ax = wmma4(aAx, bBx, ax);
  }
  for (int r = 0; r < 8; ++r) {
    int m = m0 + r + half * 8;
    float sg = gg[m] * rsqrtf(gvar[m] + BN_EPS);
    float sx = xg[m] * rsqrtf(xvar[m] + BN_EPS);
    float t = sg * (Wgb[m] - gmu[m]) + gbt[m] + sx * (Wxb[m] - xmu[m]) + xbt[m];
    float v = fmaxf(0.f, sg * ag[r] + sx * ax[r] + t);
    psi[((size_t)b * 64 + m) * 4096 + n0 + lr] = v;
  }
}

// ---------------------------------------------------------------------------
// Kernel 2: CAM attention p = softmax(rowmax - feat*feat^T)  -> [B,64,64]
// one block per batch; 8 waves cover 16 tiles (2 each), K=4096
// ---------------------------------------------------------------------------
__global__ void k_cam_att(const float* __restrict__ psi, float* __restrict__ patt) {
  __shared__ float att[64][66];
  const int tid = threadIdx.x;
  const int lane = tid & 31, half = lane >> 4, lr = lane & 15, bk = half * 2;
  const int w = tid >> 5;
  const int b = blockIdx.x;
  const float* feat = psi + (size_t)b * 64 * 4096;
  const int mt = w >> 1, dtb = (w & 1) * 2;
  const int m0 = mt * 16;
  v8f acc0 = {}, acc1 = {};
  for (int k = 0; k < 4096; k += 4) {
    v2f a = *(const v2f*)(feat + (size_t)(m0 + lr) * 4096 + k + bk);
    v2f b0 = *(const v2f*)(feat + (size_t)(dtb * 16 + lr) * 4096 + k + bk);
    v2f b1 = *(const v2f*)(feat + (size_t)((dtb + 1) * 16 + lr) * 4096 + k + bk);
    acc0 = wmma4(a, b0, acc0);
    acc1 = wmma4(a, b1, acc1);
  }
  for (int r = 0; r < 8; ++r) {
    att[m0 + r + half * 8][dtb * 16 + lr] = acc0[r];
    att[m0 + r + half * 8][(dtb + 1) * 16 + lr] = acc1[r];
  }
  __syncthreads();
  if (tid < 64) {
    // softmax(rowmax - att) == exp(rowmin - att)/sum
    float mn = att[tid][0];
    for (int j = 1; j < 64; ++j) mn = fminf(mn, att[tid][j]);
    float s = 0.f;
    for (int j = 0; j < 64; ++j) { float p = expf(mn - att[tid][j]); att[tid][j] = p; s += p; }
    float inv = 1.f / s;
    for (int j = 0; j < 64; ++j) patt[((size_t)b * 64 + tid) * 64 + j] = att[tid][j] * inv;
  }
}

// ---------------------------------------------------------------------------
// Kernel 3: cam = beta * (p @ feat) + psi   (K=64)
// ---------------------------------------------------------------------------
__global__ void k_cam_apply(const float* __restrict__ psi, const float* __restrict__ patt,
                            const float* __restrict__ beta_p, float* __restrict__ cam) {
  const int lane = threadIdx.x & 31, half = lane >> 4, lr = lane & 15, bk = half * 2;
  const int wid = blockIdx.x * 8 + (threadIdx.x >> 5);
  const int b = wid >> 10, rem = wid & 1023;
  const int nt = rem >> 2, mt = rem & 3;
  const int m0 = mt * 16, n0 = nt * 16;
  const float* feat = psi + (size_t)b * 64 * 4096;
  const float* P = patt + (size_t)b * 64 * 64;
  v8f acc = {};
  for (int k = 0; k < 64; k += 4) {
    v2f a = *(const v2f*)(P + (m0 + lr) * 64 + k + bk);
    v2f bb;
    bb.x = feat[(size_t)(k + bk) * 4096 + n0 + lr];
    bb.y = feat[(size_t)(k + bk + 1) * 4096 + n0 + lr];
    acc = wmma4(a, bb, acc);
  }
  float cb = beta_p[0];
  for (int r = 0; r < 8; ++r) {
    int m = m0 + r + half * 8;
    size_t idx = ((size_t)b * 64 + m) * 4096 + n0 + lr;
    cam[idx] = cb * acc[r] + psi[idx];
  }
}

// ---------------------------------------------------------------------------
// Kernel 4: fb/fc/fd projections. mt 0..3 -> fd rows, mt 4 -> stacked fb(0-7)/fc(8-15)
// tiles = 8*256*5 = 10240 -> 1280 blocks
// ---------------------------------------------------------------------------
__global__ void k_proj_bcd(const float* __restrict__ psi,
                           const float* __restrict__ pbw, const float* __restrict__ pbb,
                           const float* __restrict__ pcw, const float* __restrict__ pcb,
                           const float* __restrict__ pdw, const float* __restrict__ pdb,
                           float* __restrict__ fb, float* __restrict__ fc,
                           float* __restrict__ fd) {
  const int lane = threadIdx.x & 31, half = lane >> 4, lr = lane & 15, bk = half * 2;
  const int wid = blockIdx.x * 8 + (threadIdx.x >> 5);
  const int b = wid / 1280, rem = wid % 1280;
  const int nt = rem / 5, mt = rem % 5;
  const int n0 = nt * 16;
  const float* feat = psi + (size_t)b * 64 * 4096;
  const float* arow;
  if (mt < 4)       arow = pdw + (mt * 16 + lr) * 64;
  else if (lr < 8)  arow = pbw + lr * 64;
  else              arow = pcw + (lr - 8) * 64;
  v8f acc = {};
  for (int k = 0; k < 64; k += 4) {
    v2f a = *(const v2f*)(arow + k + bk);
    v2f bb;
    bb.x = feat[(size_t)(k + bk) * 4096 + n0 + lr];
    bb.y = feat[(size_t)(k + bk + 1) * 4096 + n0 + lr];
    acc = wmma4(a, bb, acc);
  }
  for (int r = 0; r < 8; ++r) {
    int m = r + half * 8;
    if (mt < 4) {
      int ch = mt * 16 + m;
      fd[((size_t)b * 64 + ch) * 4096 + n0 + lr] = acc[r] + pdb[ch];
    } else if (m < 8) {
      fb[((size_t)b * 8 + m) * 4096 + n0 + lr] = acc[r] + pbb[m];
    } else {
      fc[((size_t)b * 8 + (m - 8)) * 4096 + n0 + lr] = acc[r] + pcb[m - 8];
    }
  }
}

// ---------------------------------------------------------------------------
// Kernel 5: flash-style PAM. Q=fb^T [4096x8], K=fc [8x4096], V=fd^T [4096x64]
// Block: 64 queries, 8 waves. Iterate 64-key tiles with online softmax.
// pam = alpha * softmax(QK^T) V + psi.  512 blocks (b*64 + qtile).
// ---------------------------------------------------------------------------
__global__ void k_pam_flash(const float* __restrict__ psi,
                            const float* __restrict__ fb, const float* __restrict__ fc,
                            const float* __restrict__ fd, const float* __restrict__ alpha_p,
                            float* __restrict__ pam) {
  __shared__ float S[64][66];
  __shared__ float rscale[64];
  __shared__ float rlinv[64];
  const int tid = threadIdx.x;
  const int lane = tid & 31, half = lane >> 4, lr = lane & 15, bk = half * 2;
  const int w = tid >> 5;
  const int b = blockIdx.x >> 6;
  const int i0 = (blockIdx.x & 63) * 64;
  const int mt = w >> 1, ctb = (w & 1) * 2;
  const int m0 = mt * 16;
  const float* FB = fb + (size_t)b * 8 * 4096;
  const float* FC = fc + (size_t)b * 8 * 4096;
  const float* FD = fd + (size_t)b * 64 * 4096;

  v2f qa[2];  // Q fragment for this wave's row tile, K=8 -> 2 k-steps
  for (int ks = 0; ks < 2; ++ks) {
    int k = ks * 4;
    qa[ks].x = FB[(size_t)(k + bk) * 4096 + i0 + m0 + lr];
    qa[ks].y = FB[(size_t)(k + bk + 1) * 4096 + i0 + m0 + lr];
  }
  v8f o0 = {}, o1 = {};
  float mi = -INFINITY, li = 0.f;  // live in threads 0..63 (row = tid)

  for (int jt = 0; jt < 64; ++jt) {
    const int j0 = jt * 64;
    // S tiles for this wave (two 16x16 column tiles)
    for (int t = 0; t < 2; ++t) {
      const int d0 = (ctb + t) * 16;
      v8f s = {};
      for (int ks = 0; ks < 2; ++ks) {
        int k = ks * 4;
        v2f kb;
        kb.x = FC[(size_t)(k + bk) * 4096 + j0 + d0 + lr];
        kb.y = FC[(size_t)(k + bk + 1) * 4096 + j0 + d0 + lr];
        s = wmma4(qa[ks], kb, s);
      }
      for (int r = 0; r < 8; ++r) S[m0 + r + half * 8][d0 + lr] = s[r];
    }
    __syncthreads();
    // online softmax: one row per thread 0..63
    if (tid < 64) {
      float mnew = mi;
      for (int j = 0; j < 64; ++j) mnew = fmaxf(mnew, S[tid][j]);
      float sc = expf(mi - mnew);
      float ls = 0.f;
      for (int j = 0; j < 64; ++j) { float p = expf(S[tid][j] - mnew); S[tid][j] = p; ls += p; }
      li = li * sc + ls;
      mi = mnew;
      rscale[tid] = sc;
    }
    __syncthreads();
    // rescale O, then O += P * V^T (K=64)
    float rs[8];
    for (int r = 0; r < 8; ++r) rs[r] = rscale[m0 + r + half * 8];
    for (int r = 0; r < 8; ++r) { o0[r] *= rs[r]; o1[r] *= rs[r]; }
    for (int k = 0; k < 64; k += 4) {
      v2f a = *(const v2f*)(&S[m0 + lr][k + bk]);
      v2f v0 = *(const v2f*)(FD + (size_t)(ctb * 16 + lr) * 4096 + j0 + k + bk);
      v2f v1 = *(const v2f*)(FD + (size_t)((ctb + 1) * 16 + lr) * 4096 + j0 + k + bk);
      o0 = wmma4(a, v0, o0);
      o1 = wmma4(a, v1, o1);
    }
    __syncthreads();
  }
  if (tid < 64) rlinv[tid] = 1.f / li;
  __syncthreads();
  const float al = alpha_p[0];
  for (int r = 0; r < 8; ++r) {
    int row = m0 + r + half * 8;
    int i = i0 + row;
    float inv = rlinv[row];
    {
      size_t idx = ((size_t)b * 64 + (ctb * 16 + lr)) * 4096 + i;
      pam[idx] = al * o0[r] * inv + psi[idx];
    }
    {
      size_t idx = ((size_t)b * 64 + ((ctb + 1) * 16 + lr)) * 4096 + i;
      pam[idx] = al * o1[r] * inv + psi[idx];
    }
  }
}

// ---------------------------------------------------------------------------
// Kernel 6: da = sigmoid(BN(psi_w . (cam*pam) + psi_b))   [B,4096]
// ---------------------------------------------------------------------------
__global__ void k_gate(const float* __restrict__ cam, const float* __restrict__ pam,
                       const float* __restrict__ pw, const float* __restrict__ pb,
                       const float* __restrict__ gma, const float* __restrict__ bta,
                       const float* __restrict__ mu, const float* __restrict__ var,
                       float* __restrict__ da) {
  const int t = blockIdx.x * 256 + threadIdx.x;  // 0..32767
  const int b = t >> 12, p = t & 4095;
  const float* C = cam + (size_t)b * 64 * 4096 + p;
  const float* P = pam + (size_t)b * 64 * 4096 + p;
  float acc = pb[0];
  for (int c = 0; c < 64; ++c) acc += pw[c] * C[(size_t)c * 4096] * P[(size_t)c * 4096];
  float s = gma[0] * rsqrtf(var[0] + BN_EPS);
  float gate = s * (acc - mu[0]) + bta[0];
  da[t] = 1.f / (1.f + expf(-gate));
}

// ---------------------------------------------------------------------------
// Kernel 7: out = x * da (broadcast over 256 channels), float4 streams
// ---------------------------------------------------------------------------
__global__ void k_out(const float* __restrict__ x, const float* __restrict__ da,
                      float* __restrict__ out) {
  const size_t i = (size_t)blockIdx.x * 256 + threadIdx.x;  // float4 index
  const size_t e = i * 4;
  const int b = (int)(e >> 20);  // / (256*4096)
  const int p = (int)(e & 4095);
  float4 xv = ((const float4*)x)[i];
  const float4 dv = *(const float4*)(da + (size_t)b * 4096 + p);
  float4 ov = make_float4(xv.x * dv.x, xv.y * dv.y, xv.z * dv.z, xv.w * dv.w);
  ((float4*)out)[i] = ov;
}

// ---------------------------------------------------------------------------
extern "C" void kernel_launch(void* const* d_in, const int* in_sizes, int n_in,
                              void* d_out, int out_size, void* d_ws, size_t ws_size,
                              hipStream_t stream) {
  const float* g = (const float*)d_in[0];
  const float* x = (const float*)d_in[1];
  const float* Wg_w = (const float*)d_in[2];
  const float* Wg_b = (const float*)d_in[3];
  const float* bng_gamma = (const float*)d_in[4];
  const float* bng_beta = (const float*)d_in[5];
  const float* bng_mean = (const float*)d_in[6];
  const float* bng_var = (const float*)d_in[7];
  const float* Wx_w = (const float*)d_in[8];
  const float* Wx_b = (const float*)d_in[9];
  const float* bnx_gamma = (const float*)d_in[10];
  const float* bnx_beta = (const float*)d_in[11];
  const float* bnx_mean = (const float*)d_in[12];
  const float* bnx_var = (const float*)d_in[13];
  const float* psi_w = (const float*)d_in[14];
  const float* psi_b = (const float*)d_in[15];
  const float* bnp_gamma = (const float*)d_in[16];
  const float* bnp_beta = (const float*)d_in[17];
  const float* bnp_mean = (const float*)d_in[18];
  const float* bnp_var = (const float*)d_in[19];
  const float* pb_w = (const float*)d_in[20];
  const float* pb_b = (const float*)d_in[21];
  const float* pc_w = (const float*)d_in[22];
  const float* pc_b = (const float*)d_in[23];
  const float* pd_w = (const float*)d_in[24];
  const float* pd_b = (const float*)d_in[25];
  const float* alpha = (const float*)d_in[26];
  const float* cam_beta = (const float*)d_in[27];
  float* out = (float*)d_out;

  char* w = (char*)d_ws;
  float* psi = (float*)(w);                       // 8*64*4096 f32 = 8 MB
  float* cam = (float*)(w + (8u << 20));          // 8 MB
  float* fb  = (float*)(w + (16u << 20));         // 1 MB
  float* fc  = (float*)(w + (17u << 20));         // 1 MB
  float* fd  = (float*)(w + (18u << 20));         // 8 MB
  float* pam = (float*)(w + (26u << 20));         // 8 MB
  float* patt = (float*)(w + (34u << 20));        // 128 KB
  float* da   = (float*)(w + (34u << 20) + (1u << 17));  // 128 KB

  k_psi<<<1024, 256, 0, stream>>>(g, x, Wg_w, Wg_b, bng_gamma, bng_beta, bng_mean,
                                  bng_var, Wx_w, Wx_b, bnx_gamma, bnx_beta, bnx_mean,
                                  bnx_var, psi);
  k_cam_att<<<8, 256, 0, stream>>>(psi, patt);
  k_cam_apply<<<1024, 256, 0, stream>>>(psi, patt, cam_beta, cam);
  k_proj_bcd<<<1280, 256, 0, stream>>>(psi, pb_w, pb_b, pc_w, pc_b, pd_w, pd_b,
                                       fb, fc, fd);
  k_pam_flash<<<512, 256, 0, stream>>>(psi, fb, fc, fd, alpha, pam);
  k_gate<<<128, 256, 0, stream>>>(cam, pam, psi_w, psi_b, bnp_gamma, bnp_beta,
                                  bnp_mean, bnp_var, da);
  k_out<<<8192, 256, 0, stream>>>(x, da, out);
}